// AConnect_57449482551806
// MI455X (gfx1250) — compile-verified
//
#include <hip/hip_runtime.h>

// AConnect-style analog-mismatch forward:
//   Z[b,o] = sum_i X[b,i] * W[i,o] * Werr[b,i,o] + bias[o] * Berr[b,o]
// Memory-bound on the 512 MiB Werr stream (~23 us at 23.3 TB/s).
// WMMA f32 16x16x4 is used as a free accumulate engine: A = broadcast rows of
// X[b, i0..i0+3], B = elementwise W*Werr tile, C accumulates across K-chunks.

typedef __attribute__((ext_vector_type(2))) float v2f;
typedef __attribute__((ext_vector_type(8))) float v8f;

#define B_DIM    128
#define ROW_DIM  1024
#define OUT_DIM  1024
#define THREADS  256
#define WAVES_PER_WG (THREADS / 32)     // wave32: 8 waves
#define OTILE    (WAVES_PER_WG * 16)    // 128 outputs per workgroup

__global__ __launch_bounds__(THREADS)
void aconnect_wmma_kernel(const float* __restrict__ X,
                          const float* __restrict__ W,
                          const float* __restrict__ bias,
                          const float* __restrict__ Werr,
                          const float* __restrict__ Berr,
                          float* __restrict__ out)
{
    __shared__ float xrow[ROW_DIM];     // 4 KB: activations X[b, :]

    const int b   = blockIdx.y;
    const int tid = threadIdx.x;

    // Stage this sample's activation row into LDS (float4 per thread).
    {
        const float4* src = reinterpret_cast<const float4*>(X + (size_t)b * ROW_DIM);
        reinterpret_cast<float4*>(xrow)[tid] = src[tid];
    }
    __syncthreads();

    const int lane = tid & 31;
    const int wave = tid >> 5;
    const int half = lane >> 4;         // 0 -> this lane feeds K=0,1 ; 1 -> K=2,3
    const int n    = lane & 15;         // column within the 16-wide output tile
    const int o    = blockIdx.x * OTILE + wave * 16 + n;

    // Column pointers: row stride is OUT_DIM floats. Start at row i = 2*half.
    const float* wp = W    + (size_t)(2 * half) * OUT_DIM + o;
    const float* ep = Werr + ((size_t)b * ROW_DIM + (size_t)(2 * half)) * OUT_DIM + o;

    v8f acc = {0.f, 0.f, 0.f, 0.f, 0.f, 0.f, 0.f, 0.f};

    #pragma unroll 4
    for (int i0 = 0; i0 < ROW_DIM; i0 += 4) {
        // A(16x4): all 16 rows identical = X[b, i0..i0+3]; ds_load_b64 per lane.
        v2f a = *reinterpret_cast<const v2f*>(&xrow[i0 + 2 * half]);

        // B(4x16): memristive weight W * Werr for rows K = 2*half, 2*half+1.
        v2f bb;
        bb.x = wp[0]       * ep[0];
        bb.y = wp[OUT_DIM] * ep[OUT_DIM];

        // D = A x B + C, fp32 in / fp32 accumulate (matches reference math).
        acc = __builtin_amdgcn_wmma_f32_16x16x4_f32(false, a, false, bb,
                                                    (short)0, acc, false, false);

        // Pull the once-used Werr stream ~16 rows ahead (speculative prefetch).
        __builtin_prefetch(ep + (size_t)16 * OUT_DIM, 0, 1);

        wp += 4 * OUT_DIM;
        ep += (size_t)4 * OUT_DIM;
    }

    // C/D layout: VGPR0, lanes 0..15 hold M=0, N=lane -> the 16 tile outputs.
    if (lane < 16) {
        const size_t oi = (size_t)b * OUT_DIM + o;
        out[oi] = acc[0] + bias[o] * Berr[oi];
    }
}

extern "C" void kernel_launch(void* const* d_in, const int* in_sizes, int n_in,
                              void* d_out, int out_size, void* d_ws, size_t ws_size,
                              hipStream_t stream) {
    (void)in_sizes; (void)n_in; (void)out_size; (void)d_ws; (void)ws_size;
    const float* X    = (const float*)d_in[0];
    const float* W    = (const float*)d_in[1];
    const float* bias = (const float*)d_in[2];
    const float* Werr = (const float*)d_in[3];
    const float* Berr = (const float*)d_in[4];
    float* out = (float*)d_out;

    dim3 grid(OUT_DIM / OTILE, B_DIM);   // (8, 128) workgroups, 8 waves each
    aconnect_wmma_kernel<<<grid, THREADS, 0, stream>>>(X, W, bias, Werr, Berr, out);
}